// GM_23613730193917
// MI455X (gfx1250) — compile-verified
//
#include <hip/hip_runtime.h>
#include <math.h>

typedef __attribute__((ext_vector_type(2))) float v2f;
typedef __attribute__((ext_vector_type(8))) float v8f;

#define NPOINTS 262144
#define KCOMP   32
#define DDIM    16
#define TROW    20                       // 16 (A^T row) + 1 (-c) + 3 zero pad
#define TSZ     (KCOMP * DDIM * TROW)    // 10240 floats = 40 KB

#define WAVES_PER_BLOCK 8
#define STRIPS_PER_WAVE 4                // all 4 in flight simultaneously
#define NSTRIPS  (NPOINTS / 16)                                   // 16384
#define NBLOCKS  (NSTRIPS / (WAVES_PER_BLOCK * STRIPS_PER_WAVE))  // 512
#define TOTAL_WAVES (NBLOCKS * WAVES_PER_BLOCK)                   // 4096

// ---------------------------------------------------------------------------
// Prep kernel (1 block): builds the augmented/transposed weight matrix
//   Tg[(k*16+e)*20 + d]  = l_fac[k][d][e]        (d < 16)
//   Tg[(k*16+e)*20 + 16] = -sum_d l_fac[k,d,e]*m_w[k,d]
//   Tg[.. 17..19]        = 0
// and coef[k] = softmax(p)[k] * (2pi)^-8 * |det(l_fac_k)| / 16^8
// ---------------------------------------------------------------------------
__global__ void __launch_bounds__(256) gm_prep(const float* __restrict__ lfac,
                                               const float* __restrict__ mw,
                                               const float* __restrict__ plog,
                                               float* __restrict__ Tg,
                                               float* __restrict__ coefg)
{
  __shared__ float M[KCOMP][DDIM * DDIM + 1];   // +1 pad: dodge bank conflicts
  const int t = threadIdx.x;

  for (int idx = t; idx < KCOMP * DDIM * DDIM; idx += 256) {
    int k = idx >> 8, i = idx & 255;
    M[k][i] = lfac[idx];
  }
  __syncthreads();

  // transposed 16x16 block per component
  for (int idx = t; idx < KCOMP * DDIM * DDIM; idx += 256) {
    int k = idx >> 8, r = idx & 255;
    int e = r >> 4, d = r & 15;
    Tg[(k * DDIM + e) * TROW + d] = M[k][d * DDIM + e];
  }
  // bias column (-c) + zero padding
  for (int idx = t; idx < KCOMP * DDIM; idx += 256) {
    int k = idx >> 4, e = idx & 15;
    float s = 0.f;
    for (int d = 0; d < DDIM; ++d) s = fmaf(M[k][d * DDIM + e], mw[k * DDIM + d], s);
    float* p = Tg + (k * DDIM + e) * TROW;
    p[16] = -s; p[17] = 0.f; p[18] = 0.f; p[19] = 0.f;
  }
  __syncthreads();

  if (t < KCOMP) {
    // |det| via LU with partial pivoting (destroys M[t]; sign irrelevant)
    float* A = M[t];
    float det = 1.f;
    for (int i = 0; i < DDIM; ++i) {
      int p = i; float pv = fabsf(A[i * DDIM + i]);
      for (int r = i + 1; r < DDIM; ++r) {
        float v = fabsf(A[r * DDIM + i]);
        if (v > pv) { pv = v; p = r; }
      }
      if (p != i) {
        for (int c = i; c < DDIM; ++c) {
          float tmp = A[i * DDIM + c];
          A[i * DDIM + c] = A[p * DDIM + c];
          A[p * DDIM + c] = tmp;
        }
      }
      float piv = A[i * DDIM + i];
      det *= piv;
      float inv = 1.f / piv;
      for (int r = i + 1; r < DDIM; ++r) {
        float f = A[r * DDIM + i] * inv;
        for (int c = i + 1; c < DDIM; ++c)
          A[r * DDIM + c] = fmaf(-f, A[i * DDIM + c], A[r * DDIM + c]);
      }
    }
    // softmax over 32 logits (redundant per thread, trivial)
    float mx = plog[0];
    for (int k = 1; k < KCOMP; ++k) mx = fmaxf(mx, plog[k]);
    float sum = 0.f;
    for (int k = 0; k < KCOMP; ++k) sum += __expf(plog[k] - mx);
    float pw = __expf(plog[t] - mx) / sum;
    const float CC = 9.58520e-17f;   // (2*pi)^-8 / 16^8
    coefg[t] = pw * CC * fabsf(det);
  }
}

// ---------------------------------------------------------------------------
// Main kernel: per wave, four 16-point strips in flight. For each component j:
//   D(16x16) = A_j'^T (16x20) x X'(20x16) via 5x V_WMMA_F32_16X16X4_F32
//   (4 independent accumulator chains share the same A fragments)
//   quad[n]  = sum_e D[e][n]^2  -> 8 FMAs + one shfl_xor(16)
//   out[n]  += coef[j] * exp(-quad/32)
// ---------------------------------------------------------------------------
__global__ void __launch_bounds__(256) gm_main(const float* __restrict__ x,
                                               const float* __restrict__ Tg,
                                               const float* __restrict__ coefg,
                                               float* __restrict__ out)
{
  __shared__ float T[TSZ];
  __shared__ float coef[KCOMP];
  const int tid = threadIdx.x;
  for (int i = tid; i < TSZ; i += 256) T[i] = Tg[i];
  if (tid < KCOMP) coef[tid] = coefg[tid];
  __syncthreads();

  const int lane = tid & 31;
  const int m    = lane & 15;       // = point index n within strip (B/D cols)
  const int hi   = lane >> 4;       // which half of the wave
  const int w    = blockIdx.x * WAVES_PER_BLOCK + (tid >> 5);

  v2f bconst;                       // augmented x row: [1,0] / [0,0]
  bconst[0] = hi ? 0.f : 1.f;
  bconst[1] = 0.f;

  // load B fragments for 4 strips
  v2f b[STRIPS_PER_WAVE][5];
#pragma unroll
  for (int i = 0; i < STRIPS_PER_WAVE; ++i) {
    const int strip = w + TOTAL_WAVES * i;
    const v2f* xr = (const v2f*)(x + (strip * 16 + m) * DDIM);
#pragma unroll
    for (int s = 0; s < 4; ++s) b[i][s] = xr[2 * s + hi];
    b[i][4] = bconst;
  }

  float o[STRIPS_PER_WAVE] = {0.f, 0.f, 0.f, 0.f};

  for (int j = 0; j < KCOMP; ++j) {
    const v2f* Tp = (const v2f*)(T + (j * DDIM + m) * TROW);
    v8f acc[STRIPS_PER_WAVE];
#pragma unroll
    for (int i = 0; i < STRIPS_PER_WAVE; ++i)
      acc[i] = (v8f){0.f, 0.f, 0.f, 0.f, 0.f, 0.f, 0.f, 0.f};

#pragma unroll
    for (int s = 0; s < 5; ++s) {
      v2f af = Tp[2 * s + hi];      // (A[m][4s+2hi], A[m][4s+2hi+1])
#pragma unroll
      for (int i = 0; i < STRIPS_PER_WAVE; ++i)
        acc[i] = __builtin_amdgcn_wmma_f32_16x16x4_f32(
            false, af, false, b[i][s], (short)0, acc[i], false, false);
    }

    const float cj = coef[j];
#pragma unroll
    for (int i = 0; i < STRIPS_PER_WAVE; ++i) {
      float ss = acc[i][0] * acc[i][0];
#pragma unroll
      for (int v = 1; v < 8; ++v) ss = fmaf(acc[i][v], acc[i][v], ss);
      ss += __shfl_xor(ss, 16, 32);           // combine e halves -> full quad
      o[i] = fmaf(cj, __expf(-0.03125f * ss), o[i]);   // exp(-quad_raw/32)
    }
  }

  if (lane < 16) {
#pragma unroll
    for (int i = 0; i < STRIPS_PER_WAVE; ++i)
      out[(w + TOTAL_WAVES * i) * 16 + lane] = o[i];
  }
}

// ---------------------------------------------------------------------------
extern "C" void kernel_launch(void* const* d_in, const int* in_sizes, int n_in,
                              void* d_out, int out_size, void* d_ws, size_t ws_size,
                              hipStream_t stream)
{
  const float* x    = (const float*)d_in[0];   // (N,16)
  const float* mw   = (const float*)d_in[1];   // (32,16)
  const float* lfac = (const float*)d_in[2];   // (32,16,16)
  const float* plog = (const float*)d_in[3];   // (32,)
  float* out = (float*)d_out;                  // (N,1)

  float* Tg    = (float*)d_ws;                 // 10240 floats
  float* coefg = Tg + TSZ;                     // 32 floats

  gm_prep<<<1, 256, 0, stream>>>(lfac, mw, plog, Tg, coefg);
  gm_main<<<NBLOCKS, 256, 0, stream>>>(x, Tg, coefg, out);
}